// UNet_14156212208355
// MI455X (gfx1250) — compile-verified
//
#include <hip/hip_runtime.h>
#include <cstdint>
#include <cstddef>

// ============================================================================
// Spherical UNet forward for MI455X (gfx1250).
// - fp32 end-to-end; channel-mixing contraction runs on V_WMMA_F32_16X16X4_F32
//   (the K=4 step == the 4 differential-operator stack per input channel).
// - A panel staged per-block into LDS via global_load_async_to_lds_b128
//   (ASYNCcnt) and read back with ds_load_b64; B streams are branchless
//   per-lane global streams, software-pipelined one channel ahead so WMMAs
//   overlap the in-flight loads (staggered s_wait_loadcnt, no full drain).
// ============================================================================

typedef float v2f __attribute__((ext_vector_type(2)));
typedef float v8f __attribute__((ext_vector_type(8)));

// ---------------------------------------------------------------------------
// Gather kernels
// ---------------------------------------------------------------------------

// out[bc, v] = sum_k in[bc, idx[v,k]] * val[v,k]     (Laplacian: K=7)
__global__ void gather_mad_kernel(const float* __restrict__ in, int ldin,
                                  const int* __restrict__ idx,
                                  const float* __restrict__ val,
                                  int K, int V, float* __restrict__ out)
{
  int v  = blockIdx.x * blockDim.x + threadIdx.x;
  int bc = blockIdx.y;
  if (v >= V) return;
  const float* src = in + (size_t)bc * ldin;
  float acc = 0.f;
  for (int k = 0; k < K; ++k) {
    int j = idx[(size_t)v * K + k];
    acc = fmaf(src[j], val[(size_t)v * K + k], acc);
  }
  out[(size_t)bc * V + v] = acc;
}

// per-face gradient (3 verts, 3 spatial comps) dotted with EW/NS tangents
__global__ void grad_tan_kernel(const float* __restrict__ in, int ldin,
                                const int* __restrict__ Gidx,
                                const float* __restrict__ Gval,
                                const float* __restrict__ EW,
                                const float* __restrict__ NS,
                                int F, float* __restrict__ outEW,
                                float* __restrict__ outNS)
{
  int f  = blockIdx.x * blockDim.x + threadIdx.x;
  int bc = blockIdx.y;
  if (f >= F) return;
  const float* src = in + (size_t)bc * ldin;
  float x0 = src[Gidx[3 * f + 0]];
  float x1 = src[Gidx[3 * f + 1]];
  float x2 = src[Gidx[3 * f + 2]];
  float ew = 0.f, ns = 0.f;
  for (int j = 0; j < 3; ++j) {
    const float* gv = Gval + ((size_t)f * 3 + j) * 3;
    float g = fmaf(gv[0], x0, fmaf(gv[1], x1, gv[2] * x2));
    ew = fmaf(g, EW[3 * f + j], ew);
    ns = fmaf(g, NS[3 * f + j], ns);
  }
  outEW[(size_t)bc * F + f] = ew;
  outNS[(size_t)bc * F + f] = ns;
}

// face -> vertex averaging for both tangent fields (K=6)
__global__ void f2v_kernel(const float* __restrict__ inEW,
                           const float* __restrict__ inNS, int F,
                           const int* __restrict__ idx,
                           const float* __restrict__ val,
                           int V, float* __restrict__ outEW,
                           float* __restrict__ outNS)
{
  int v  = blockIdx.x * blockDim.x + threadIdx.x;
  int bc = blockIdx.y;
  if (v >= V) return;
  const float* se = inEW + (size_t)bc * F;
  const float* sn = inNS + (size_t)bc * F;
  float ae = 0.f, an = 0.f;
  for (int k = 0; k < 6; ++k) {
    int   j = idx[(size_t)v * 6 + k];
    float w = val[(size_t)v * 6 + k];
    ae = fmaf(se[j], w, ae);
    an = fmaf(sn[j], w, an);
  }
  outEW[(size_t)bc * V + v] = ae;
  outNS[(size_t)bc * V + v] = an;
}

// ---------------------------------------------------------------------------
// Channel-mixing GEMM on V_WMMA_F32_16X16X4_F32.
//   out[b, o_off+o, v] = bias[o] + sum_c sum_{k<4} coeffs[o,c,k] * op_k[b,c,v]
// A fragment (16x4, MxK): lanes 0-15 -> K=0,1 ; lanes 16-31 -> K=2,3
// B fragment (4x16, KxN): lanes 0-15 -> rows 0,1 ; lanes 16-31 -> rows 2,3
// D (16x16):  VGPR r: M=r (lanes 0-15), M=r+8 (lanes 16-31); N = lane&15
//
// A panel (16 rows x 64-channel chunk x 4 coeffs = 16KB) is staged into LDS
// once per block via global_load_async_to_lds_b128; all clamping is
// index-clamping (garbage rows/cols never reach a stored output element).
// B streams are double-buffered: loads for channel cc+1 are issued before the
// WMMAs of channel cc, forcing distinct registers and staggered loadcnt waits.
// ---------------------------------------------------------------------------
#define GEMM_NT    4    // 16-wide N sub-tiles per wave
#define GEMM_WAVES 4    // waves per block
#define CCHUNK     64   // channels of A panel staged per LDS round

__global__ __launch_bounds__(GEMM_WAVES * 32)
void meshconv_gemm_wmma(const float* __restrict__ op0, int ld0,
                        const float* __restrict__ op1,
                        const float* __restrict__ op2,
                        const float* __restrict__ op3,
                        const float* __restrict__ coeffs,
                        const float* __restrict__ bias,
                        int C, int O, int V,
                        float* __restrict__ out, int Ocat, int o_off)
{
  // Single shared array -> sits at LDS byte offset 0 (used by the async asm).
  __shared__ float alds[16 * CCHUNK * 4];

  const int  lane = threadIdx.x & 31;
  const int  wave = threadIdx.x >> 5;
  const int  ncol = lane & 15;
  const bool hi   = lane >= 16;
  const int  o0   = blockIdx.y * 16;
  const int  b    = blockIdx.z;
  const int  vbase = (blockIdx.x * GEMM_WAVES + wave) * (16 * GEMM_NT);

  v8f acc[GEMM_NT] = {};

  // ---- branchless per-lane B streams --------------------------------------
  // bf[0] row: lanes<16 -> op0 (stride ld0), lanes>=16 -> op2 (stride V)
  // bf[1] row: lanes<16 -> op1 (stride V),   lanes>=16 -> op3 (stride V)
  const float* q0 = hi ? (op2 + (size_t)b * C * V) : (op0 + (size_t)b * C * ld0);
  const float* q1 = hi ? (op3 + (size_t)b * C * V) : (op1 + (size_t)b * C * V);
  const size_t s0 = hi ? (size_t)V : (size_t)ld0;
  const size_t s1 = (size_t)V;

  int nn[GEMM_NT];
#pragma unroll
  for (int t = 0; t < GEMM_NT; ++t) {
    int n = vbase + t * 16 + ncol;
    nn[t] = (n < V) ? n : (V - 1);           // clamp: cols >= V never stored
  }

  const int ksel = hi ? 2 : 0;               // which K pair this half holds

  for (int c0 = 0; c0 < C; c0 += CCHUNK) {
    const int chunk = (C - c0 < CCHUNK) ? (C - c0) : CCHUNK;

    __syncthreads();   // previous chunk's LDS reads complete
    // ---- cooperative async stage of A panel: 16 rows x chunk x 16B --------
    for (int t = threadIdx.x; t < 16 * chunk; t += GEMM_WAVES * 32) {
      int row = t / chunk;
      int cc  = t - row * chunk;
      int rowg = o0 + row;
      if (rowg >= O) rowg = O - 1;           // clamp: rows >= O never stored
      const float*   gp = coeffs + (((size_t)rowg * C + (c0 + cc)) << 2);
      const uint32_t lo = (uint32_t)((row * CCHUNK + cc) * 16);
      asm volatile("global_load_async_to_lds_b128 %0, %1, off"
                   :: "v"(lo), "v"(gp) : "memory");
    }
    asm volatile("s_wait_asynccnt 0x0" ::: "memory");
    __syncthreads();   // A panel visible to all waves

    // ---- software-pipelined channel loop ----------------------------------
    v2f bf[GEMM_NT];
#pragma unroll
    for (int t = 0; t < GEMM_NT; ++t) {      // prologue: channel c0's B row
      bf[t][0] = q0[nn[t]];
      bf[t][1] = q1[nn[t]];
    }
    for (int cc = 0; cc < chunk; ++cc) {
      v2f cur[GEMM_NT];
#pragma unroll
      for (int t = 0; t < GEMM_NT; ++t) cur[t] = bf[t];
      // issue next channel's loads (uniform no-op advance on last iteration)
      const size_t adv0 = (cc + 1 < chunk) ? s0 : 0;
      const size_t adv1 = (cc + 1 < chunk) ? s1 : 0;
      q0 += adv0;
      q1 += adv1;
#pragma unroll
      for (int t = 0; t < GEMM_NT; ++t) {
        bf[t][0] = q0[nn[t]];
        bf[t][1] = q1[nn[t]];
      }
      // A fragment from LDS (ds_load_b64)
      v2f a = *(const v2f*)&alds[(ncol * CCHUNK + cc) * 4 + ksel];
#pragma unroll
      for (int t = 0; t < GEMM_NT; ++t)
        acc[t] = __builtin_amdgcn_wmma_f32_16x16x4_f32(
            false, a, false, cur[t], (short)0, acc[t], false, false);
    }
    q0 += s0;          // step to the next chunk's first channel
    q1 += s1;
  }

#pragma unroll
  for (int t = 0; t < GEMM_NT; ++t) {
    int n = vbase + t * 16 + ncol;
    if (n >= V) continue;
#pragma unroll
    for (int r = 0; r < 8; ++r) {
      int o = o0 + r + (hi ? 8 : 0);
      if (o < O)
        out[((size_t)b * Ocat + (o_off + o)) * V + n] = acc[t][r] + bias[o];
    }
  }
}

// ---------------------------------------------------------------------------
// Training-mode BatchNorm over (batch, vertex) + ReLU
// ---------------------------------------------------------------------------
__global__ void bn_stats_kernel(const float* __restrict__ x, int V, int B,
                                int C, float* __restrict__ stats)
{
  const int c = blockIdx.x;
  const int tid = threadIdx.x;
  float s = 0.f, ss = 0.f;
  for (int b = 0; b < B; ++b) {
    const float* row = x + ((size_t)b * C + c) * V;
    for (int v = tid; v < V; v += blockDim.x) {
      float val = row[v];
      s  += val;
      ss += val * val;
    }
  }
  __shared__ float sh0[256];
  __shared__ float sh1[256];
  sh0[tid] = s;
  sh1[tid] = ss;
  __syncthreads();
  for (int off = 128; off > 0; off >>= 1) {
    if (tid < off) { sh0[tid] += sh0[tid + off]; sh1[tid] += sh1[tid + off]; }
    __syncthreads();
  }
  if (tid == 0) {
    float inv  = 1.f / (float)(B * V);
    float mean = sh0[0] * inv;
    float var  = sh1[0] * inv - mean * mean;
    stats[2 * c]     = mean;
    stats[2 * c + 1] = var;
  }
}

__global__ void bn_relu_kernel(float* __restrict__ x,
                               const float* __restrict__ stats,
                               const float* __restrict__ g,
                               const float* __restrict__ bb,
                               int C, int V)
{
  int v  = blockIdx.x * blockDim.x + threadIdx.x;
  int bc = blockIdx.y;
  if (v >= V) return;
  int c = bc % C;
  float mean = stats[2 * c], var = stats[2 * c + 1];
  float r = rsqrtf(var + 1e-5f);
  size_t i = (size_t)bc * V + v;
  float val = fmaf((x[i] - mean) * r, g[c], bb[c]);
  x[i] = fmaxf(val, 0.f);
}

// ---------------------------------------------------------------------------
// Plumbing: zero-pad along vertex dim; copy channels into concat buffer
// ---------------------------------------------------------------------------
__global__ void pad_kernel(const float* __restrict__ in, int Vin,
                           float* __restrict__ out, int Vout)
{
  int v  = blockIdx.x * blockDim.x + threadIdx.x;
  int bc = blockIdx.y;
  if (v >= Vout) return;
  out[(size_t)bc * Vout + v] = (v < Vin) ? in[(size_t)bc * Vin + v] : 0.f;
}

__global__ void copy_ch_kernel(const float* __restrict__ in, int ldin, int Cin,
                               float* __restrict__ out, int Cout, int coff,
                               int V)
{
  int v  = blockIdx.x * blockDim.x + threadIdx.x;
  int bc = blockIdx.y;
  if (v >= V) return;
  int b = bc / Cin, c = bc - b * Cin;
  out[((size_t)b * Cout + coff + c) * V + v] =
      in[((size_t)b * Cin + c) * ldin + v];
}

// ---------------------------------------------------------------------------
// Host-side orchestration
// ---------------------------------------------------------------------------
namespace {

struct ConvP { const float* co; const float* bi; };
struct DCP   { ConvP c1; const float* g1; const float* b1;
               ConvP c2; const float* g2; const float* b2; };
struct MeshP { const int* L_idx; const float* L_val;
               const int* G_idx; const float* G_val;
               const float* EW; const float* NS;
               const int* F2V_idx; const float* F2V_val; };
struct Scr   { float* LAP; float* EWF; float* NSF;
               float* GVE; float* GVN; float* STATS; };

inline dim3 g1d(int n, int y) {
  return dim3((unsigned)((n + 255) / 256), (unsigned)y, 1);
}

void run_mesh_conv(hipStream_t s, const MeshP& m, int B, int V, int F,
                   const float* x, int ldx, int C, const ConvP& cv, int O,
                   float* out, int Ocat, int coff, const Scr& sc)
{
  dim3 bt(256, 1, 1);
  gather_mad_kernel<<<g1d(V, B * C), bt, 0, s>>>(x, ldx, m.L_idx, m.L_val, 7,
                                                 V, sc.LAP);
  grad_tan_kernel<<<g1d(F, B * C), bt, 0, s>>>(x, ldx, m.G_idx, m.G_val, m.EW,
                                               m.NS, F, sc.EWF, sc.NSF);
  f2v_kernel<<<g1d(V, B * C), bt, 0, s>>>(sc.EWF, sc.NSF, F, m.F2V_idx,
                                          m.F2V_val, V, sc.GVE, sc.GVN);
  dim3 gg((unsigned)((V + 255) / 256), (unsigned)((O + 15) / 16), (unsigned)B);
  meshconv_gemm_wmma<<<gg, dim3(GEMM_WAVES * 32, 1, 1), 0, s>>>(
      x, ldx, sc.LAP, sc.GVE, sc.GVN, cv.co, cv.bi, C, O, V, out, Ocat, coff);
}

void run_bn_relu(hipStream_t s, float* t, int B, int C, int V,
                 const float* g, const float* b, float* stats)
{
  bn_stats_kernel<<<dim3((unsigned)C, 1, 1), dim3(256, 1, 1), 0, s>>>(t, V, B,
                                                                      C, stats);
  bn_relu_kernel<<<g1d(V, B * C), dim3(256, 1, 1), 0, s>>>(t, stats, g, b, C,
                                                           V);
}

void run_dc(hipStream_t s, const MeshP& m, int B, int V, int F,
            const float* in, int ldin, int cin, int cout, const DCP& p,
            float* out, float* tmp, const Scr& sc)
{
  run_mesh_conv(s, m, B, V, F, in, ldin, cin, p.c1, cout, tmp, cout, 0, sc);
  run_bn_relu(s, tmp, B, cout, V, p.g1, p.b1, sc.STATS);
  run_mesh_conv(s, m, B, V, F, tmp, V, cout, p.c2, cout, out, cout, 0, sc);
  run_bn_relu(s, out, B, cout, V, p.g2, p.b2, sc.STATS);
}

} // namespace

extern "C" void kernel_launch(void* const* d_in, const int* in_sizes, int n_in,
                              void* d_out, int out_size, void* d_ws,
                              size_t ws_size, hipStream_t stream)
{
  (void)in_sizes; (void)n_in; (void)out_size; (void)ws_size;

  const int B = 4;
  int NV[7], NF[7];
  for (int l = 0; l <= 6; ++l) {
    NV[l] = 10 * (1 << (2 * l)) + 2;
    NF[l] = 20 * (1 << (2 * l));
  }

  // ---- unpack inputs (depth-first, dict insertion order): 169 tensors ----
  int p = 0;
  const float* x_in = (const float*)d_in[p++];                       // x
  ConvP in_conv  = { (const float*)d_in[p], (const float*)d_in[p + 1] }; p += 2;
  ConvP out_conv = { (const float*)d_in[p], (const float*)d_in[p + 1] }; p += 2;
  DCP down[6];
  for (int i = 0; i < 6; ++i) {
    down[i].c1 = { (const float*)d_in[p + 0], (const float*)d_in[p + 1] };
    down[i].g1 = (const float*)d_in[p + 2];
    down[i].b1 = (const float*)d_in[p + 3];
    down[i].c2 = { (const float*)d_in[p + 4], (const float*)d_in[p + 5] };
    down[i].g2 = (const float*)d_in[p + 6];
    down[i].b2 = (const float*)d_in[p + 7];
    p += 8;
  }
  ConvP upc[6];
  DCP   updc[6];
  for (int i = 0; i < 6; ++i) {
    upc[i]     = { (const float*)d_in[p + 0], (const float*)d_in[p + 1] };
    updc[i].c1 = { (const float*)d_in[p + 2], (const float*)d_in[p + 3] };
    updc[i].g1 = (const float*)d_in[p + 4];
    updc[i].b1 = (const float*)d_in[p + 5];
    updc[i].c2 = { (const float*)d_in[p + 6], (const float*)d_in[p + 7] };
    updc[i].g2 = (const float*)d_in[p + 8];
    updc[i].b2 = (const float*)d_in[p + 9];
    p += 10;
  }
  MeshP mesh[7];
  for (int l = 0; l <= 6; ++l) {
    mesh[l].L_idx   = (const int*)d_in[p + 0];
    mesh[l].L_val   = (const float*)d_in[p + 1];
    mesh[l].G_idx   = (const int*)d_in[p + 2];
    mesh[l].G_val   = (const float*)d_in[p + 3];
    mesh[l].EW      = (const float*)d_in[p + 4];
    mesh[l].NS      = (const float*)d_in[p + 5];
    mesh[l].F2V_idx = (const int*)d_in[p + 6];
    mesh[l].F2V_val = (const float*)d_in[p + 7];
    p += 8;
  }

  // ---- workspace bump allocator ----
  char*  ws  = (char*)d_ws;
  size_t off = 0;
  auto alloc = [&](size_t nfloat) -> float* {
    float* r = (float*)(ws + off);
    off += (nfloat * sizeof(float) + 255) & ~(size_t)255;
    return r;
  };

  const int xsC[7] = {16, 32, 64, 128, 256, 512, 512};
  float* xs[7];
  for (int j = 0; j <= 6; ++j) xs[j] = alloc((size_t)B * xsC[j] * NV[6 - j]);

  const size_t MAXV = 6u  * 1024 * 1024;   // >= 4*32*40962 = 5.25M floats
  const size_t MAXF = 11u * 1024 * 1024;   // >= 4*32*81920 = 10.5M floats
  float* ACT1 = alloc(MAXV);
  float* ACT2 = alloc(MAXV);
  float* ACT3 = alloc(MAXV);
  float* PADB = alloc(MAXV);
  float* CAT  = alloc(MAXV);
  Scr sc;
  sc.LAP   = alloc(MAXV);
  sc.GVE   = alloc(MAXV);
  sc.GVN   = alloc(MAXV);
  sc.EWF   = alloc(MAXF);
  sc.NSF   = alloc(MAXF);
  sc.STATS = alloc(2048);

  // ---- in_conv at level 6 (no BN) ----
  run_mesh_conv(stream, mesh[6], B, NV[6], NF[6], x_in, NV[6], 4, in_conv, 16,
                xs[0], 16, 0, sc);

  // ---- encoder: truncate to coarser mesh, double conv ----
  const int dnin[6]  = {16, 32, 64, 128, 256, 512};
  const int dnout[6] = {32, 64, 128, 256, 512, 512};
  const float* cur = xs[0];
  int curLd = NV[6];
  for (int i = 0; i < 6; ++i) {
    int lvl = 5 - i;
    run_dc(stream, mesh[lvl], B, NV[lvl], NF[lvl], cur, curLd, dnin[i],
           dnout[i], down[i], xs[i + 1], ACT1, sc);
    cur = xs[i + 1];
    curLd = NV[lvl];
  }

  // ---- decoder: pad, up conv, skip concat, double conv ----
  const int upCu[6]  = {512, 256, 128, 64, 32, 16};
  const int upOut[6] = {256, 128, 64, 32, 16, 16};
  const float* h = xs[6];
  int hC = 512, hV = NV[0];
  for (int i = 0; i < 6; ++i) {
    int lvl = i + 1, V = NV[lvl], F = NF[lvl];
    int Cu = hC;          // up-conv channels (== upCu[i])
    int Cs = upCu[i];     // skip channels == Cu
    pad_kernel<<<g1d(V, B * Cu), dim3(256, 1, 1), 0, stream>>>(h, hV, PADB, V);
    // up-conv writes directly into concat buffer at channel offset Cs
    run_mesh_conv(stream, mesh[lvl], B, V, F, PADB, V, Cu, upc[i], Cu, CAT,
                  Cs + Cu, Cs, sc);
    copy_ch_kernel<<<g1d(V, B * Cs), dim3(256, 1, 1), 0, stream>>>(
        xs[5 - i], V, Cs, CAT, Cs + Cu, 0, V);
    float* ob = (i & 1) ? ACT3 : ACT2;
    run_dc(stream, mesh[lvl], B, V, F, CAT, V, Cs + Cu, upOut[i], updc[i], ob,
           ACT1, sc);
    h = ob;
    hC = upOut[i];
    hV = V;
  }

  // ---- out_conv at level 6 -> d_out [4, 8, 40962] f32 ----
  run_mesh_conv(stream, mesh[6], B, NV[6], NF[6], h, NV[6], 16, out_conv, 8,
                (float*)d_out, 8, 0, sc);
}